// AUX_SSALoss_64579128263170
// MI455X (gfx1250) — compile-verified
//
#include <hip/hip_runtime.h>
#include <hip/hip_bf16.h>
#include <math.h>

// SSA loss, algebraically collapsed:
//   corr_diff(m,n)^2 = sum_b ( s_m^2 + s_n^2 - 2*d_mn^2 )
// where q_m = ||x_m[b]||^2, p_mn = <x_m[b], x_n[b]>,
//       s_m = q_m / max(sqrt(q_m),eps)^2,  d_mn = p_mn/(max(sqrt(q_m),eps)*max(sqrt(q_n),eps)).
// Per-row dots computed as diagonals of 16x16 Gram tiles via V_WMMA_F32_16X16X4_F32.

typedef __attribute__((ext_vector_type(2))) float v2f;
typedef __attribute__((ext_vector_type(8))) float v8f;

#define B_ROWS 128
#define DIMS   512

__global__ __launch_bounds__(192)
void ssa_stats_kernel(const float* __restrict__ x0,
                      const float* __restrict__ x1,
                      const float* __restrict__ x2,
                      float* __restrict__ stats)   // stats[c*128 + row], c in {q0,q1,q2,p01,p02,p12}
{
    __shared__ float lds[6][32][8];

    const int tid  = threadIdx.x;
    const int w    = tid >> 5;     // wave id = combo id, uniform per wave
    const int l    = tid & 31;     // lane
    const int tile = blockIdx.x;   // 16-row tile, 0..7

    // combo table: m = {0,1,2,0,0,1}[w], n = {0,1,2,1,2,2}[w]
    const int mi = (w == 1 || w == 5) ? 1 : ((w == 2) ? 2 : 0);
    const int ni = (w == 0) ? 0 : ((w == 1 || w == 3) ? 1 : 2);
    const float* Xm = (mi == 0) ? x0 : ((mi == 1) ? x1 : x2);
    const float* Xn = (ni == 0) ? x0 : ((ni == 1) ? x1 : x2);

    // fp32 16x4 A layout (ISA 7.12.2): lanes 0-15 row M=l hold K=k0,k0+1 in v0,v1;
    // lanes 16-31 row M=l-16 hold K=k0+2,k0+3. B (4x16) mirrors this per lane with
    // N = lane&15, so the per-lane load pattern is identical for A and B.
    const int r    = l & 15;
    const int kofs = (l < 16) ? 0 : 2;
    const float* arow = Xm + (size_t)(tile * 16 + r) * DIMS + kofs;
    const float* brow = Xn + (size_t)(tile * 16 + r) * DIMS + kofs;

    v8f acc = {0.f, 0.f, 0.f, 0.f, 0.f, 0.f, 0.f, 0.f};

    #pragma unroll 4
    for (int k0 = 0; k0 < DIMS; k0 += 4) {
        v2f a = *(const v2f*)(arow + k0);
        v2f b = *(const v2f*)(brow + k0);
        acc = __builtin_amdgcn_wmma_f32_16x16x4_f32(
            /*neg_a=*/false, a, /*neg_b=*/false, b,
            /*c_mod=*/(short)0, acc, /*reuse_a=*/false, /*reuse_b=*/false);
    }

    // Spill the 16x16 tile to LDS, then pull out the diagonal (per-row dot products).
    #pragma unroll
    for (int i = 0; i < 8; ++i) lds[w][l][i] = acc[i];
    __syncthreads();

    if (l < 16) {
        // diag element j: VGPR j&7, lane j (j<8) or lane j+16 (j>=8)
        const int vg   = l & 7;
        const int lane = (l < 8) ? l : (l + 16);
        stats[w * B_ROWS + tile * 16 + l] = lds[w][lane][vg];
    }
}

__global__ __launch_bounds__(32)
void ssa_finalize_kernel(const float* __restrict__ stats,
                         const float* __restrict__ la,
                         const float* __restrict__ lv,
                         const float* __restrict__ lt,
                         float* __restrict__ out)
{
    const int l = threadIdx.x;   // single wave32
    float S01 = 0.f, S02 = 0.f, S12 = 0.f;

    for (int b = l; b < B_ROWS; b += 32) {
        const float q0  = stats[0 * B_ROWS + b];
        const float q1  = stats[1 * B_ROWS + b];
        const float q2  = stats[2 * B_ROWS + b];
        const float p01 = stats[3 * B_ROWS + b];
        const float p02 = stats[4 * B_ROWS + b];
        const float p12 = stats[5 * B_ROWS + b];

        const float d0 = fmaxf(sqrtf(q0), 1e-12f);
        const float d1 = fmaxf(sqrtf(q1), 1e-12f);
        const float d2 = fmaxf(sqrtf(q2), 1e-12f);

        const float s0 = q0 / (d0 * d0);
        const float s1 = q1 / (d1 * d1);
        const float s2 = q2 / (d2 * d2);

        const float c01 = p01 / (d0 * d1);
        const float c02 = p02 / (d0 * d2);
        const float c12 = p12 / (d1 * d2);

        S01 += s0 * s0 + s1 * s1 - 2.f * c01 * c01;
        S02 += s0 * s0 + s2 * s2 - 2.f * c02 * c02;
        S12 += s1 * s1 + s2 * s2 - 2.f * c12 * c12;
    }

    // wave32 butterfly reduction
    #pragma unroll
    for (int off = 16; off > 0; off >>= 1) {
        S01 += __shfl_xor(S01, off, 32);
        S02 += __shfl_xor(S02, off, 32);
        S12 += __shfl_xor(S12, off, 32);
    }

    if (l == 0) {
        const float L[3] = {la[0], lv[0], lt[0]};
        float cd[3][3];
        const float r01 = sqrtf(1e-12f + fmaxf(S01, 0.f));
        const float r02 = sqrtf(1e-12f + fmaxf(S02, 0.f));
        const float r12 = sqrtf(1e-12f + fmaxf(S12, 0.f));
        cd[0][0] = cd[1][1] = cd[2][2] = sqrtf(1e-12f);  // multiplied by focal(0)=0 anyway
        cd[0][1] = cd[1][0] = r01;
        cd[0][2] = cd[2][0] = r02;
        cd[1][2] = cd[2][1] = r12;

        #pragma unroll
        for (int m = 0; m < 3; ++m) {
            float acc = 0.f;
            #pragma unroll
            for (int n = 0; n < 3; ++n) {
                const float d = L[m] - L[n];
                const float f = (d > 1e-12f) ? logf(0.5f * d + 1.0f) : 0.f;
                acc += f * cd[m][n];
            }
            out[m] = acc;
        }
    }
}

extern "C" void kernel_launch(void* const* d_in, const int* in_sizes, int n_in,
                              void* d_out, int out_size, void* d_ws, size_t ws_size,
                              hipStream_t stream) {
    const float* x0 = (const float*)d_in[0];
    const float* x1 = (const float*)d_in[1];
    const float* x2 = (const float*)d_in[2];
    const float* la = (const float*)d_in[3];
    const float* lv = (const float*)d_in[4];
    const float* lt = (const float*)d_in[5];
    float* stats = (float*)d_ws;     // 6*128 floats = 3 KB
    float* out   = (float*)d_out;

    ssa_stats_kernel<<<dim3(B_ROWS / 16), dim3(192), 0, stream>>>(x0, x1, x2, stats);
    ssa_finalize_kernel<<<dim3(1), dim3(32), 0, stream>>>(stats, la, lv, lt, out);
}